// ImprovedSEALModel_53953379173089
// MI455X (gfx1250) — compile-verified
//
#include <hip/hip_runtime.h>
#include <hip/hip_bf16.h>

// Problem constants (match reference)
#define NN 100000
#define EE 1600000
#define GG 2000
#define LL 2000
#define EPS 1e-5f

typedef __attribute__((ext_vector_type(2))) float v2f;
typedef __attribute__((ext_vector_type(8))) float v8f;

static __device__ __forceinline__ void atomic_add_f32(float* p, float v) {
    __hip_atomic_fetch_add(p, v, __ATOMIC_RELAXED, __HIP_MEMORY_SCOPE_AGENT);
}

// ---------------------------------------------------------------------------
// Degree: deg[n] = sum over edges with dst==n, then dis[n]=rsqrt(deg+1)
// ---------------------------------------------------------------------------
__global__ void deg_accum_kernel(const int* __restrict__ dst, float* __restrict__ deg, int e) {
    int t = blockIdx.x * blockDim.x + threadIdx.x;
    if (t < e) atomic_add_f32(&deg[dst[t]], 1.0f);
}

__global__ void deg_finish_kernel(float* __restrict__ deg, int n) {
    int t = blockIdx.x * blockDim.x + threadIdx.x;
    if (t < n) deg[t] = rsqrtf(deg[t] + 1.0f);
}

// ---------------------------------------------------------------------------
// Fused dual GEMM with native fp32 WMMA (V_WMMA_F32_16X16X4_F32):
//   hlin = A @ W           (needed for edge gather)
//   agg  = hlin*dis^2 + b + (A @ Wr) + br    (self-loop + bias + residual init)
// One wave (32 lanes) per 16x16 output tile. A operand shared by both WMMAs.
// Requires nrows % 16 == 0 (N = 100000 = 6250*16), FOUT % 16 == 0, FIN % 4 == 0.
// ---------------------------------------------------------------------------
template<int FIN, int FOUT>
__global__ __launch_bounds__(32)
void gcn_gemm_kernel(const float* __restrict__ A,
                     const float* __restrict__ W,  const float* __restrict__ bias,
                     const float* __restrict__ Wr, const float* __restrict__ brias,
                     const float* __restrict__ dis,
                     float* __restrict__ hlin, float* __restrict__ agg) {
    const int lane    = threadIdx.x;       // 0..31
    const int rowTile = blockIdx.x;        // N/16
    const int colTile = blockIdx.y;        // FOUT/16
    const int r15     = lane & 15;
    const int hh      = lane >> 4;         // 0 or 1
    const int row     = rowTile * 16 + r15;   // A-matrix row held by this lane
    const int col     = colTile * 16 + r15;   // B/D column held by this lane

    const float* __restrict__ arow = A + (size_t)row * FIN;

    v8f accW = {};
    v8f accR = {};
    #pragma unroll 8
    for (int k0 = 0; k0 < FIN; k0 += 4) {
        const int ka = k0 + 2 * hh;        // lanes 0-15: K=k0,k0+1 ; lanes 16-31: K=k0+2,k0+3
        v2f a;  a.x  = arow[ka];
                a.y  = arow[ka + 1];
        v2f bw; bw.x = W [(size_t)ka * FOUT + col];
                bw.y = W [(size_t)(ka + 1) * FOUT + col];
        v2f bb; bb.x = Wr[(size_t)ka * FOUT + col];
                bb.y = Wr[(size_t)(ka + 1) * FOUT + col];
        accW = __builtin_amdgcn_wmma_f32_16x16x4_f32(false, a, false, bw, (short)0, accW, false, false);
        accR = __builtin_amdgcn_wmma_f32_16x16x4_f32(false, a, false, bb, (short)0, accR, false, false);
    }

    const float bc = bias[col] + brias[col];
    #pragma unroll
    for (int r = 0; r < 8; ++r) {
        const int m  = rowTile * 16 + r + 8 * hh;  // C/D: VGPR r holds M=r (lanes 0-15), M=r+8 (16-31)
        const float d  = dis[m];
        const float hv = accW[r];
        hlin[(size_t)m * FOUT + col] = hv;
        agg [(size_t)m * FOUT + col] = hv * d * d + bc + accR[r];
    }
}

// ---------------------------------------------------------------------------
// Edge scatter: agg[dst] += hlin[src] * dis[src]*dis[dst]. float4 per thread.
// ---------------------------------------------------------------------------
template<int F>
__global__ void gcn_scatter_kernel(const float* __restrict__ hlin,
                                   const int* __restrict__ src, const int* __restrict__ dst,
                                   const float* __restrict__ dis, float* __restrict__ agg,
                                   long long total) {
    constexpr int C = F / 4;
    long long t = (long long)blockIdx.x * blockDim.x + threadIdx.x;
    if (t >= total) return;
    const int e = (int)(t / C);
    const int c = (int)(t % C);
    const int s = src[e];
    const int d = dst[e];
    const float coef = dis[s] * dis[d];
    const float4 v = *(const float4*)(hlin + (size_t)s * F + 4 * c);
    float* p = agg + (size_t)d * F + 4 * c;
    atomic_add_f32(p + 0, v.x * coef);
    atomic_add_f32(p + 1, v.y * coef);
    atomic_add_f32(p + 2, v.z * coef);
    atomic_add_f32(p + 3, v.w * coef);
}

// ---------------------------------------------------------------------------
// LayerNorm + ReLU, one thread per node.
// ---------------------------------------------------------------------------
template<int F>
__global__ void gcn_ln_relu_kernel(const float* __restrict__ agg,
                                   const float* __restrict__ g, const float* __restrict__ beta,
                                   float* __restrict__ out, int n) {
    int node = blockIdx.x * blockDim.x + threadIdx.x;
    if (node >= n) return;
    float v[F];
    const float* p = agg + (size_t)node * F;
    float m = 0.0f;
    #pragma unroll
    for (int f = 0; f < F; ++f) { v[f] = p[f]; m += v[f]; }
    m *= (1.0f / F);
    float var = 0.0f;
    #pragma unroll
    for (int f = 0; f < F; ++f) { float d = v[f] - m; var += d * d; }
    var *= (1.0f / F);
    const float rs = rsqrtf(var + EPS);
    float* q = out + (size_t)node * F;
    #pragma unroll
    for (int f = 0; f < F; ++f) {
        float y = (v[f] - m) * rs * g[f] + beta[f];
        q[f] = y > 0.0f ? y : 0.0f;
    }
}

// ---------------------------------------------------------------------------
// Global mean pool over subgraphs
// ---------------------------------------------------------------------------
__global__ void pool_cnt_kernel(const int* __restrict__ batch, float* __restrict__ cnt, int n) {
    int t = blockIdx.x * blockDim.x + threadIdx.x;
    if (t < n) atomic_add_f32(&cnt[batch[t]], 1.0f);
}

__global__ void pool_sum_kernel(const float* __restrict__ h, const int* __restrict__ batch,
                                float* __restrict__ gemb, int n) {
    int t = blockIdx.x * blockDim.x + threadIdx.x;  // over n*16
    if (t >= n * 16) return;
    const int node = t >> 4;
    const int f    = t & 15;
    atomic_add_f32(&gemb[(size_t)batch[node] * 16 + f], h[(size_t)node * 16 + f]);
}

__global__ void pool_div_kernel(float* __restrict__ gemb, const float* __restrict__ cnt, int g16) {
    int t = blockIdx.x * blockDim.x + threadIdx.x;
    if (t < g16) gemb[t] /= fmaxf(cnt[t >> 4], 1.0f);
}

// ---------------------------------------------------------------------------
// Link-prediction MLP: [32]->relu 16->relu 8->1->sigmoid. One thread per link.
// ---------------------------------------------------------------------------
__global__ void mlp_link_kernel(const float* __restrict__ gemb, const int* __restrict__ link,
                                const float* __restrict__ Wm1, const float* __restrict__ bm1,
                                const float* __restrict__ Wm2, const float* __restrict__ bm2,
                                const float* __restrict__ Wm3, const float* __restrict__ bm3,
                                float* __restrict__ out, int nl) {
    int l = blockIdx.x * blockDim.x + threadIdx.x;
    if (l >= nl) return;
    const int ga = link[l];
    const int gb = link[nl + l];
    float feat[32];
    #pragma unroll
    for (int f = 0; f < 16; ++f) feat[f]      = gemb[(size_t)ga * 16 + f];
    #pragma unroll
    for (int f = 0; f < 16; ++f) feat[16 + f] = gemb[(size_t)gb * 16 + f];
    float z1[16];
    #pragma unroll
    for (int o = 0; o < 16; ++o) {
        float s = bm1[o];
        #pragma unroll
        for (int i = 0; i < 32; ++i) s += feat[i] * Wm1[i * 16 + o];
        z1[o] = fmaxf(s, 0.0f);
    }
    float z2[8];
    #pragma unroll
    for (int o = 0; o < 8; ++o) {
        float s = bm2[o];
        #pragma unroll
        for (int i = 0; i < 16; ++i) s += z1[i] * Wm2[i * 8 + o];
        z2[o] = fmaxf(s, 0.0f);
    }
    float s = bm3[0];
    #pragma unroll
    for (int i = 0; i < 8; ++i) s += z2[i] * Wm3[i];
    out[l] = 1.0f / (1.0f + expf(-s));
}

// ---------------------------------------------------------------------------
// Launch. Workspace layout (floats):
//   bufA N*64 | bufB N*32 | hlin N*64 | agg N*64 | dis N | cnt G | gemb G*16
//   total ~= 90.2 MB
// ---------------------------------------------------------------------------
extern "C" void kernel_launch(void* const* d_in, const int* in_sizes, int n_in,
                              void* d_out, int out_size, void* d_ws, size_t ws_size,
                              hipStream_t stream) {
    const float* x     = (const float*)d_in[0];
    const int*   src   = (const int*)d_in[1];
    const int*   dst   = (const int*)d_in[2];
    const int*   batch = (const int*)d_in[3];
    const int*   link  = (const int*)d_in[4];

    const float *W[4], *b[4], *Wr[4], *br[4], *g[4], *bt[4];
    for (int i = 0; i < 4; ++i) {
        const int base = 5 + 6 * i;
        W[i]  = (const float*)d_in[base + 0];
        b[i]  = (const float*)d_in[base + 1];
        Wr[i] = (const float*)d_in[base + 2];
        br[i] = (const float*)d_in[base + 3];
        g[i]  = (const float*)d_in[base + 4];
        bt[i] = (const float*)d_in[base + 5];
    }
    const float* Wm1 = (const float*)d_in[29];
    const float* bm1 = (const float*)d_in[30];
    const float* Wm2 = (const float*)d_in[31];
    const float* bm2 = (const float*)d_in[32];
    const float* Wm3 = (const float*)d_in[33];
    const float* bm3 = (const float*)d_in[34];
    float* out = (float*)d_out;

    float* ws = (float*)d_ws;
    size_t o = 0;
    float* bufA = ws + o; o += (size_t)NN * 64;
    float* bufB = ws + o; o += (size_t)NN * 32;
    float* hlin = ws + o; o += (size_t)NN * 64;
    float* agg  = ws + o; o += (size_t)NN * 64;
    float* dis  = ws + o; o += (size_t)NN;
    float* cnt  = ws + o; o += (size_t)GG;
    float* gemb = ws + o; o += (size_t)GG * 16;

    hipMemsetAsync(dis,  0, (size_t)NN * sizeof(float), stream);
    hipMemsetAsync(cnt,  0, (size_t)GG * sizeof(float), stream);
    hipMemsetAsync(gemb, 0, (size_t)GG * 16 * sizeof(float), stream);

    // degrees
    deg_accum_kernel<<<(EE + 255) / 256, 256, 0, stream>>>(dst, dis, EE);
    deg_finish_kernel<<<(NN + 255) / 256, 256, 0, stream>>>(dis, NN);

    const int rowTiles = NN / 16;  // 6250

    // ---- layer 1: 128 -> 64 ----
    gcn_gemm_kernel<128, 64><<<dim3(rowTiles, 4), 32, 0, stream>>>(
        x, W[0], b[0], Wr[0], br[0], dis, hlin, agg);
    {
        long long total = (long long)EE * (64 / 4);
        gcn_scatter_kernel<64><<<(unsigned)((total + 255) / 256), 256, 0, stream>>>(
            hlin, src, dst, dis, agg, total);
    }
    gcn_ln_relu_kernel<64><<<(NN + 127) / 128, 128, 0, stream>>>(agg, g[0], bt[0], bufA, NN);

    // ---- layer 2: 64 -> 32 ----
    gcn_gemm_kernel<64, 32><<<dim3(rowTiles, 2), 32, 0, stream>>>(
        bufA, W[1], b[1], Wr[1], br[1], dis, hlin, agg);
    {
        long long total = (long long)EE * (32 / 4);
        gcn_scatter_kernel<32><<<(unsigned)((total + 255) / 256), 256, 0, stream>>>(
            hlin, src, dst, dis, agg, total);
    }
    gcn_ln_relu_kernel<32><<<(NN + 127) / 128, 128, 0, stream>>>(agg, g[1], bt[1], bufB, NN);

    // ---- layer 3: 32 -> 16 ----
    gcn_gemm_kernel<32, 16><<<dim3(rowTiles, 1), 32, 0, stream>>>(
        bufB, W[2], b[2], Wr[2], br[2], dis, hlin, agg);
    {
        long long total = (long long)EE * (16 / 4);
        gcn_scatter_kernel<16><<<(unsigned)((total + 255) / 256), 256, 0, stream>>>(
            hlin, src, dst, dis, agg, total);
    }
    gcn_ln_relu_kernel<16><<<(NN + 127) / 128, 128, 0, stream>>>(agg, g[2], bt[2], bufA, NN);

    // ---- layer 4: 16 -> 16 ----
    gcn_gemm_kernel<16, 16><<<dim3(rowTiles, 1), 32, 0, stream>>>(
        bufA, W[3], b[3], Wr[3], br[3], dis, hlin, agg);
    {
        long long total = (long long)EE * (16 / 4);
        gcn_scatter_kernel<16><<<(unsigned)((total + 255) / 256), 256, 0, stream>>>(
            hlin, src, dst, dis, agg, total);
    }
    gcn_ln_relu_kernel<16><<<(NN + 127) / 128, 128, 0, stream>>>(agg, g[3], bt[3], bufB, NN);

    // ---- pooling + MLP ----
    pool_cnt_kernel<<<(NN + 255) / 256, 256, 0, stream>>>(batch, cnt, NN);
    pool_sum_kernel<<<(NN * 16 + 255) / 256, 256, 0, stream>>>(bufB, batch, gemb, NN);
    pool_div_kernel<<<(GG * 16 + 255) / 256, 256, 0, stream>>>(gemb, cnt, GG * 16);
    mlp_link_kernel<<<(LL + 63) / 64, 64, 0, stream>>>(
        gemb, link, Wm1, bm1, Wm2, bm2, Wm3, bm3, out, LL);
}